// DepthWiseConvCaps_12137577578719
// MI455X (gfx1250) — compile-verified
//
#include <hip/hip_runtime.h>
#include <hip/hip_bf16.h>

// Fused DepthWiseConvCaps + EM-routing for MI455X (gfx1250, wave32).
//
// Strategy (memory-roofline driven):
//  - One 288-thread block (9 waves) per output site (32*15*15 = 7200 sites).
//  - Phase 1: pose x weight 4x4x4 matmuls via V_WMMA_F32_16X16X4_F32.
//    Each WMMA packs 4 capsules' poses as A (16x4) and the same 4 capsules'
//    weights as B (4x16); the four diagonal 4x4 blocks of D are the valid
//    v-tiles. All loaded data is used; v never touches HBM (stays in LDS).
//  - Phase 2: 3 EM iterations in LDS. Thread t = k*32 + c, so each wave owns
//    one routing position k and lanes = capsules c -> softmax over capsules
//    is a pure wave32 shuffle reduction.
//  - HBM traffic ~ read x (67MB) + a (4MB) + write out (16MB) ~= 4us floor
//    at 23.3 TB/s; avoids ~265MB of v materialization traffic.

typedef __attribute__((ext_vector_type(2))) float v2f;
typedef __attribute__((ext_vector_type(8))) float v8f;

#define LN_2PI 1.8378770664093453f

__global__ __launch_bounds__(288) void emcaps_kernel(
    const float* __restrict__ x,       // (32,32,32,32,16)
    const float* __restrict__ a,       // (32,32,32,32,1)
    const float* __restrict__ w,       // (9,32,4,4)
    const float* __restrict__ beta_u,  // (32,)
    const float* __restrict__ beta_a,  // (32,)
    float* __restrict__ out)           // mu: 7200*512, then a_out: 7200*32
{
    __shared__ float v_lds[9*32*16];   // 18 KB: v[k][c][p*4+r]
    __shared__ float a_lds[9*32];      // a_in[k][c]
    __shared__ float r_lds[9*32];      // routing r[k][c]
    __shared__ float coeff_lds[9*32];  // r / (r_sum + eps)
    __shared__ float mu_lds[32*16];    // mu[c][s]
    __shared__ float i2s_lds[32*16];   // 0.5 / sigma_sq
    __shared__ float lsg_lds[32*16];   // 0.5 * log(sigma_sq)
    __shared__ float aout_lds[32];

    const int t    = threadIdx.x;      // 0..287
    const int lane = t & 31;
    const int wave = t >> 5;           // 0..8
    const int site = blockIdx.x;       // 0..7199
    const int bb   = site / 225;
    const int rem  = site % 225;
    const int oy   = rem / 15;
    const int ox   = rem % 15;

    // ---- stage a_in: one element per thread, (k,c) = (t/32, t%32)
    {
        int k = t >> 5, c = t & 31;
        int ky = k / 3, kx = k % 3;
        int y = oy*2 + ky, xx = ox*2 + kx;
        a_lds[t] = a[(((size_t)bb*32 + y)*32 + xx)*32 + c];
    }

    // ---- Phase 1: v = pose x W via V_WMMA_F32_16X16X4_F32
    // 72 tiles (tile = k*8 + cgroup); each of the 9 waves runs 8 tiles.
    {
        const int lhalf = lane & 15;
        const int qb    = (lane >> 4) << 1;  // K-pair: 0 or 2 (A/B VGPR layout)
        const int csub  = lhalf >> 2;        // local capsule within group of 4
        const int pr    = lhalf & 3;         // p (A rows) / r (B cols)
        for (int i = 0; i < 8; ++i) {
            const int tile = wave*8 + i;
            const int k  = tile >> 3;        // 0..8 kernel position
            const int cg = tile & 7;         // capsule group (4 caps each)
            const int ky = k/3, kx = k%3;
            const int y = oy*2 + ky, xx = ox*2 + kx;
            const int cA = cg*4 + csub;
            // A 16x4 f32: lane L<16 holds row M=L, {K=0,1}; L>=16 holds {K=2,3}
            const float* xp = x + ((((size_t)bb*32 + y)*32 + xx)*32 + cA)*16
                                + pr*4 + qb;
            v2f A; A.x = xp[0]; A.y = xp[1];
            // B 4x16 f32: lane L<16 holds col N=L, {K=0,1}; L>=16 holds {K=2,3}
            const float* wp = w + (((k*32 + cA)*4 + qb)*4) + pr;
            v2f B; B.x = wp[0]; B.y = wp[4];
            v8f Cz = {};
            v8f D = __builtin_amdgcn_wmma_f32_16x16x4_f32(
                        false, A, false, B, (short)0, Cz, false, false);
            // Valid diagonal blocks: D[4cc+p][4cc+r].
            // C/D layout: VGPR i = row i (lanes 0-15) / row 8+i (lanes 16-31).
            // => rows 0..7 live in lanes 0..7 (cc=0,1); rows 8..15 in lanes
            //    24..31 (cc=2,3). Element index = 4*(cc&1)+p in both cases.
            // Per-element ternaries -> 4 v_cndmask (avoid dynamic extracts).
            if (lane < 8 || lane >= 24) {
                const int cc    = (lane < 8) ? (lane >> 2) : ((lane - 16) >> 2);
                const bool hi   = (cc & 1);
                const float e0  = hi ? D[4] : D[0];
                const float e1  = hi ? D[5] : D[1];
                const float e2  = hi ? D[6] : D[2];
                const float e3  = hi ? D[7] : D[3];
                const int cdiag = cg*4 + cc;
                float* vd = &v_lds[(k*32 + cdiag)*16 + (lane & 3)];
                vd[0] = e0; vd[4] = e1; vd[8] = e2; vd[12] = e3;
            }
        }
    }
    __syncthreads();

    // ---- Phase 2: EM routing, 3 iterations. t = k*32 + c.
    const int c = lane;                 // capsule
    const float bu = beta_u[c];
    const float ba = beta_a[c];

    for (int it = 0; it < 3; ++it) {
        float r_val;
        if (it == 0) {
            r_val = a_lds[t] * (1.0f/32.0f);          // a_in / C
        } else {
            // ln_ap[k][c] = sum_s ln_pjh + log(a_out[c])
            float lnap = __logf(aout_lds[c]);
            const float* vp = &v_lds[t*16];
            #pragma unroll
            for (int s = 0; s < 16; ++s) {
                float d = vp[s] - mu_lds[c*16 + s];
                lnap += -(d*d)*i2s_lds[c*16 + s] - lsg_lds[c*16 + s]
                        - 0.5f*LN_2PI;
            }
            // softmax over capsules = over lanes of this wave (wave32)
            float m = lnap;
            for (int o = 16; o; o >>= 1) m = fmaxf(m, __shfl_xor(m, o, 32));
            float e = __expf(lnap - m);
            float ss = e;
            for (int o = 16; o; o >>= 1) ss += __shfl_xor(ss, o, 32);
            r_val = e / ss;
        }
        r_lds[t] = r_val;
        __syncthreads();

        // coeff = r / (sum_k r + eps)
        float rs = 0.f;
        #pragma unroll
        for (int kk = 0; kk < 9; ++kk) rs += r_lds[kk*32 + c];
        coeff_lds[t] = r_val / (rs + 1e-6f);
        __syncthreads();

        // M-step over 512 (c2,s2) pairs
        for (int idx = t; idx < 512; idx += 288) {
            const int c2 = idx >> 4, s2 = idx & 15;
            float cf[9], vv[9];
            float mu_v = 0.f;
            #pragma unroll
            for (int kk = 0; kk < 9; ++kk) {
                cf[kk] = coeff_lds[kk*32 + c2];
                vv[kk] = v_lds[(kk*32 + c2)*16 + s2];
                mu_v += cf[kk]*vv[kk];
            }
            float sg = 1e-6f;
            #pragma unroll
            for (int kk = 0; kk < 9; ++kk) {
                float d = vv[kk] - mu_v;
                sg += cf[kk]*d*d;
            }
            mu_lds[idx]  = mu_v;
            i2s_lds[idx] = 0.5f / sg;           // for -(v-mu)^2/(2*sigma)
            lsg_lds[idx] = 0.5f * __logf(sg);   // log(sqrt(sigma))
        }
        __syncthreads();

        // a_out[c] = sigmoid(lam*(beta_a - r_sum*(16*beta_u + sum_s logsig)))
        if (t < 32) {
            float rs_c = 0.f;
            #pragma unroll
            for (int kk = 0; kk < 9; ++kk) rs_c += r_lds[kk*32 + t];
            float lsum = 0.f;
            #pragma unroll
            for (int s = 0; s < 16; ++s) lsum += lsg_lds[t*16 + s];
            float cost = rs_c * (16.f*bu + lsum);
            float z = 0.001f * (ba - cost);
            aout_lds[t] = 1.f / (1.f + __expf(-z));
        }
        __syncthreads();
    }

    // ---- outputs: mu (site*512) then a_out (offset 7200*512, site*32)
    for (int idx = t; idx < 512; idx += 288)
        out[(size_t)site*512 + idx] = mu_lds[idx];
    if (t < 32)
        out[3686400 + (size_t)site*32 + t] = aout_lds[t];
}

extern "C" void kernel_launch(void* const* d_in, const int* in_sizes, int n_in,
                              void* d_out, int out_size, void* d_ws, size_t ws_size,
                              hipStream_t stream) {
    const float* x  = (const float*)d_in[0];
    const float* a  = (const float*)d_in[1];
    const float* w  = (const float*)d_in[2];
    const float* bu = (const float*)d_in[3];
    const float* ba = (const float*)d_in[4];
    float* out = (float*)d_out;
    // 7200 sites (32 batches x 15 x 15), 288 threads = 9 wave32 per site
    emcaps_kernel<<<dim3(7200), dim3(288), 0, stream>>>(x, a, w, bu, ba, out);
}